// ProofGNN_4337916969137
// MI455X (gfx1250) — compile-verified
//
#include <hip/hip_runtime.h>
#include <hip/hip_bf16.h>

typedef float v2f __attribute__((ext_vector_type(2)));
typedef float v8f __attribute__((ext_vector_type(8)));

// ---------------------------------------------------------------------------
// Kernel 0: init degree accumulator to 1 (self-loop), stored as uint in acc.
// ---------------------------------------------------------------------------
__global__ void gcn_init_deg(unsigned* __restrict__ accu, int n) {
    int v = blockIdx.x * blockDim.x + threadIdx.x;
    if (v < n) accu[v] = 1u;
}

// ---------------------------------------------------------------------------
// Kernel 1: degree count over edges (4 edges per thread, b128 loads).
// ---------------------------------------------------------------------------
__global__ void gcn_degree(const int* __restrict__ dst, unsigned* __restrict__ accu, int E) {
    int t = blockIdx.x * blockDim.x + threadIdx.x;
    int e = t * 4;
    if (e + 3 < E) {
        int4 d = *(const int4*)(dst + e);
        atomicAdd(&accu[d.x], 1u);
        atomicAdd(&accu[d.y], 1u);
        atomicAdd(&accu[d.z], 1u);
        atomicAdd(&accu[d.w], 1u);
    } else {
        for (; e < E; ++e) atomicAdd(&accu[dst[e]], 1u);
    }
}

// ---------------------------------------------------------------------------
// Kernel 2: dinv = rsqrt(deg); valn = x * dinv; acc <- 0.0f (bits 0u).
// acc buffer read as uint, re-zeroed through the same uint pointer
// (0u is bit-identical to 0.0f) to avoid strict-aliasing reordering.
// ---------------------------------------------------------------------------
__global__ void gcn_node1(unsigned* __restrict__ accu, const float* __restrict__ x,
                          float* __restrict__ dinv, float* __restrict__ valn, int n) {
    int v = blockIdx.x * blockDim.x + threadIdx.x;
    if (v < n) {
        float d  = (float)accu[v];
        float di = rsqrtf(d);          // deg >= 1 always (self loop)
        dinv[v] = di;
        valn[v] = x[v] * di;
        accu[v] = 0u;                  // == 0.0f for the float scatter pass
    }
}

// ---------------------------------------------------------------------------
// Kernel 3 / 5: scatter-add valn[src] into acc[dst] (4 edges per thread).
// ---------------------------------------------------------------------------
__global__ void gcn_scatter(const int* __restrict__ src, const int* __restrict__ dst,
                            const float* __restrict__ val, float* __restrict__ acc, int E) {
    int t = blockIdx.x * blockDim.x + threadIdx.x;
    int e = t * 4;
    if (e + 3 < E) {
        int4 s = *(const int4*)(src + e);
        int4 d = *(const int4*)(dst + e);
        atomicAdd(&acc[d.x], val[s.x]);
        atomicAdd(&acc[d.y], val[s.y]);
        atomicAdd(&acc[d.z], val[s.z]);
        atomicAdd(&acc[d.w], val[s.w]);
    } else {
        for (; e < E; ++e) atomicAdd(&acc[dst[e]], val[src[e]]);
    }
}

// ---------------------------------------------------------------------------
// Kernel 4: per-node fused  s1 -> relu(W1*s1+b1) @ W2  via WMMA f32 16x16x4.
// One wave handles a tile of 16 nodes. A operand (documented 16x4 f32 layout):
//   VGPR0: lanes 0-15 -> K=4c+0, lanes 16-31 -> K=4c+2
//   VGPR1: lanes 0-15 -> K=4c+1, lanes 16-31 -> K=4c+3,  M = lane%16
// W2 is folded into A; B = all-ones (layout invariant) so D[m][n] = h2[m] for
// every column n. C/D layout: c[r] = h2[r + 8*(lane>=16)] on every lane.
// Writes valn[v] = h2[v]*dinv[v] and re-zeroes acc for the second scatter.
// ---------------------------------------------------------------------------
__global__ void gcn_node2_wmma(const float* __restrict__ dinv, float* __restrict__ acc,
                               float* __restrict__ valn,
                               const float* __restrict__ W1, const float* __restrict__ b1,
                               const float* __restrict__ W2, int n) {
    int lane = threadIdx.x & 31;
    int wave = (int)((blockIdx.x * blockDim.x + threadIdx.x) >> 5);
    int base = wave * 16;
    if (base >= n) return;                 // uniform per wave: EXEC stays all-1s

    int m  = lane & 15;                    // node-in-tile for the A operand
    int v  = base + m;
    // s1 = dinv*acc1 + x*dinv^2  (valn currently holds x*dinv)
    float s1 = dinv[v] * (acc[v] + valn[v]);

    int hi = lane >> 4;                    // 0: lanes 0-15, 1: lanes 16-31
    v8f c = {};
    v2f b; b.x = 1.0f; b.y = 1.0f;
#pragma unroll
    for (int ch = 0; ch < 4; ++ch) {
        int k0 = ch * 4 + hi * 2;
        int k1 = k0 + 1;
        v2f a;
        a.x = fmaxf(fmaf(W1[k0], s1, b1[k0]), 0.0f) * W2[k0];
        a.y = fmaxf(fmaf(W1[k1], s1, b1[k1]), 0.0f) * W2[k1];
        // 8 args: (neg_a, A, neg_b, B, c_mod, C, reuse_a, reuse_b)
        c = __builtin_amdgcn_wmma_f32_16x16x4_f32(false, a, false, b,
                                                  (short)0, c, false, false);
    }

    // Every lane 0-15 holds h2[base+0..7] in c[0..7]; lanes 16-31 hold
    // h2[base+8..15]. Let lane 0 and lane 16 each write their 8 nodes.
    if (m == 0) {
        int nb = base + hi * 8;
#pragma unroll
        for (int r = 0; r < 8; ++r) {
            int u = nb + r;
            valn[u] = c[r] * dinv[u];      // h2 * dinv for second propagation
        }
    }
    if (lane < 16) acc[base + lane] = 0.0f;
}

// ---------------------------------------------------------------------------
// Kernel 6: out[v] = dinv[v]*(acc2[v] + h2[v]*dinv[v]) + b2
// ---------------------------------------------------------------------------
__global__ void gcn_out(const float* __restrict__ dinv, const float* __restrict__ acc,
                        const float* __restrict__ valn, const float* __restrict__ b2,
                        float* __restrict__ out, int n) {
    int v = blockIdx.x * blockDim.x + threadIdx.x;
    if (v < n) out[v] = fmaf(dinv[v], acc[v] + valn[v], b2[0]);
}

extern "C" void kernel_launch(void* const* d_in, const int* in_sizes, int n_in,
                              void* d_out, int out_size, void* d_ws, size_t ws_size,
                              hipStream_t stream) {
    const float* x    = (const float*)d_in[0];
    const int*   ei   = (const int*)d_in[1];   // JAX x64-off: int32 on device
    const float* W1   = (const float*)d_in[2];
    const float* b1   = (const float*)d_in[3];
    const float* W2   = (const float*)d_in[4];
    const float* b2   = (const float*)d_in[5];
    float*       out  = (float*)d_out;

    const int N = in_sizes[0];          // 1,000,000 nodes
    const int E = in_sizes[1] / 2;      // 5,000,000 edges
    const int* src = ei;
    const int* dst = ei + E;

    // Workspace: 3 float arrays of N (12 MB total)
    float* acc  = (float*)d_ws;
    float* dinv = acc + N;
    float* valn = dinv + N;

    const int BT = 256;
    const int gN  = (N + BT - 1) / BT;
    const int gE4 = ((E + 3) / 4 + BT - 1) / BT;
    // node2: 16 nodes per wave, 8 waves per 256-thread block -> 128 nodes/block
    const int gW  = ((N + 15) / 16 + 7) / 8;

    gcn_init_deg<<<gN, BT, 0, stream>>>((unsigned*)acc, N);
    gcn_degree  <<<gE4, BT, 0, stream>>>(dst, (unsigned*)acc, E);
    gcn_node1   <<<gN, BT, 0, stream>>>((unsigned*)acc, x, dinv, valn, N);
    gcn_scatter <<<gE4, BT, 0, stream>>>(src, dst, valn, acc, E);
    gcn_node2_wmma<<<gW, BT, 0, stream>>>(dinv, acc, valn, W1, b1, W2, N);
    gcn_scatter <<<gE4, BT, 0, stream>>>(src, dst, valn, acc, E);
    gcn_out     <<<gN, BT, 0, stream>>>(dinv, acc, valn, b2, out, N);
}